// MT2DFD_Torch_89867895701505
// MI455X (gfx1250) — compile-verified
//
#include <hip/hip_runtime.h>
#include <math.h>

// ---------------------------------------------------------------------------
// MT 2-D TE forward model on MI455X (gfx1250).
// Block-tridiagonal Thomas solver: 39 blocks of 49x49 (padded to 64x64),
// entirely LDS-resident, one workgroup per frequency.
// All dense O(n^3) work (blocked-LU trailing updates + multi-RHS TRSM panel
// updates) in complex-f32 via V_WMMA_F32_16X16X4_F32; serial recurrences /
// boundary / receiver math in f64 scalar VALU.
// ---------------------------------------------------------------------------

#define NZC 50          // z cells
#define NYC 40          // y cells
#define NZI 49          // interior z nodes  (block size)
#define NYI 39          // interior y nodes  (number of blocks)
#define NN  (NZI * NYI) // 1911 unknowns
#define PADN 64         // padded block size (4 x 16 WMMA tiles)
#define STR  65         // LDS row stride (floats) - avoids 64-bank conflicts
#define NTHREADS 256    // 8 wave32

static constexpr double PI_D  = 3.14159265358979323846;
static constexpr double MIU_D = 4e-7 * PI_D;

struct dc { double re, im; };
__device__ __forceinline__ dc cmul(dc a, dc b){ return {a.re*b.re - a.im*b.im, a.re*b.im + a.im*b.re}; }
__device__ __forceinline__ dc cadd(dc a, dc b){ return {a.re+b.re, a.im+b.im}; }
__device__ __forceinline__ dc csub(dc a, dc b){ return {a.re-b.re, a.im-b.im}; }
__device__ __forceinline__ dc cdiv(dc a, dc b){
  double d = 1.0/(b.re*b.re + b.im*b.im);
  return {(a.re*b.re + a.im*b.im)*d, (a.im*b.re - a.re*b.im)*d};
}

typedef float v8f __attribute__((ext_vector_type(8)));
typedef float v2f __attribute__((ext_vector_type(2)));

// ---------------------------------------------------------------------------
// One complex 16x16 <- 16x16 * 16x16 tile update:  D -= A*B   (all f32)
// K=16 accumulated as 4 chunks of V_WMMA_F32_16X16X4_F32 (4 real WMMAs per
// chunk for the complex product; imag-A negated in VALU since f32 WMMA has
// no A/B NEG modifier per ISA 7.12).
// Must be called with a full wave (EXEC all ones): callers branch only on
// wave-uniform tile indices. A/B/D may point into the same LDS array as long
// as the written D tile is element-disjoint from the A/B tiles (true for all
// call sites: trailing-update / panel-update tiles never overlap the factor
// panels they consume).
// ---------------------------------------------------------------------------
__device__ __forceinline__ void cgemm16_sub(
    const float* Are, const float* Aim, int I0, int K0,
    const float* Bre, const float* Bim, int KB0, int J0,
    float* Dre, float* Dim, int DI0, int DJ0, int lane)
{
#if defined(__gfx1250__)
  v8f cr = {0.f,0.f,0.f,0.f,0.f,0.f,0.f,0.f};
  v8f ci = {0.f,0.f,0.f,0.f,0.f,0.f,0.f,0.f};
  const int mrow = lane & 15;          // A row / B,C,D column within tile
  const int base = (lane >> 4) << 1;   // K sub-offset per ISA A-16x4 layout
#pragma unroll
  for (int kk = 0; kk < 4; ++kk) {
    const int ka = K0  + 4*kk + base;
    const int kb = KB0 + 4*kk + base;
    v2f ar, ai, nai, br, bi;
    ar.x = Are[(I0+mrow)*STR + ka];       ar.y = Are[(I0+mrow)*STR + ka + 1];
    ai.x = Aim[(I0+mrow)*STR + ka];       ai.y = Aim[(I0+mrow)*STR + ka + 1];
    br.x = Bre[(kb  )*STR + J0 + mrow];   br.y = Bre[(kb+1)*STR + J0 + mrow];
    bi.x = Bim[(kb  )*STR + J0 + mrow];   bi.y = Bim[(kb+1)*STR + J0 + mrow];
    nai.x = -ai.x; nai.y = -ai.y;
    cr = __builtin_amdgcn_wmma_f32_16x16x4_f32(false, ar,  false, br, (short)0, cr, false, false);
    cr = __builtin_amdgcn_wmma_f32_16x16x4_f32(false, nai, false, bi, (short)0, cr, false, false);
    ci = __builtin_amdgcn_wmma_f32_16x16x4_f32(false, ar,  false, bi, (short)0, ci, false, false);
    ci = __builtin_amdgcn_wmma_f32_16x16x4_f32(false, ai,  false, br, (short)0, ci, false, false);
  }
  const int hi = lane >> 4;
  const int n  = lane & 15;
#pragma unroll
  for (int r = 0; r < 8; ++r) {
    const int M = r + 8*hi;              // C/D VGPR layout per ISA 7.12.2
    Dre[(DI0+M)*STR + DJ0 + n] -= cr[r];
    Dim[(DI0+M)*STR + DJ0 + n] -= ci[r];
  }
#else
  // non-gfx1250 / host pass fallback (never executed on MI455X)
  for (int e = lane; e < 256; e += 32) {
    int M = e >> 4, n = e & 15;
    float accr = 0.f, acci = 0.f;
    for (int k = 0; k < 16; ++k) {
      float arx = Are[(I0+M)*STR + K0 + k],  aix = Aim[(I0+M)*STR + K0 + k];
      float brx = Bre[(KB0+k)*STR + J0 + n], bix = Bim[(KB0+k)*STR + J0 + n];
      accr += arx*brx - aix*bix;
      acci += arx*bix + aix*brx;
    }
    Dre[(DI0+M)*STR + DJ0 + n] -= accr;
    Dim[(DI0+M)*STR + DJ0 + n] -= acci;
  }
#endif
}

// ---------------------------------------------------------------------------
// 1-D TE Thomas solve (complex f64, tridiagonal 49x49) for the left/right
// boundary columns. Executed by a single lane; E must hold NZC+1 entries.
// ---------------------------------------------------------------------------
__device__ void solve_1d_te(const double* dzs, const double* hzs,
                            const double* __restrict__ sig, int col,
                            double om_miu, dc* E)
{
  dc cp[NZI], dp[NZI];
  {
    double sn = (sig[0*NYC+col]*dzs[0] + sig[1*NYC+col]*dzs[1]) / (dzs[0] + dzs[1]);
    dc m0  = { -(1.0/dzs[0] + 1.0/dzs[1]), om_miu * sn * hzs[0] };
    dc inv0 = cdiv({1.0, 0.0}, m0);
    double off0 = 1.0/dzs[1];
    cp[0] = { inv0.re*off0, inv0.im*off0 };
    dc b0 = { -1.0/dzs[0], 0.0 };
    dp[0] = cmul(b0, inv0);
  }
  for (int i = 1; i < NZI; ++i) {
    double sn = (sig[i*NYC+col]*dzs[i] + sig[(i+1)*NYC+col]*dzs[i+1]) / (dzs[i] + dzs[i+1]);
    dc mi = { -(1.0/dzs[i] + 1.0/dzs[i+1]), om_miu * sn * hzs[i] };
    double offm1 = 1.0/dzs[i];                       // sub-diagonal
    dc den = csub(mi, {offm1*cp[i-1].re, offm1*cp[i-1].im});
    dc inv = cdiv({1.0, 0.0}, den);
    if (i < NZI-1) { double offi = 1.0/dzs[i+1]; cp[i] = {inv.re*offi, inv.im*offi}; }
    dc t = { -offm1*dp[i-1].re, -offm1*dp[i-1].im }; // rhs zero for i>0
    dp[i] = cmul(t, inv);
  }
  dc x = dp[NZI-1];
  E[NZI] = x;
  for (int i = NZI-2; i >= 0; --i) { x = csub(dp[i], cmul(cp[i], x)); E[1+i] = x; }
  E[0]   = {1.0, 0.0};
  E[NZC] = {0.0, 0.0};
}

// ---------------------------------------------------------------------------
// Dynamic-LDS layout (≈132.7 KB of the 320 KB WGP LDS)
// ---------------------------------------------------------------------------
static constexpr size_t SMEM_BYTES =
    (size_t)(51 + 51 + 41 + 41 + 41 + NN + NN) * sizeof(dc)   // complex f64 arrays
  + (size_t)(NZC + NYC + NZI + NYI) * sizeof(double)          // geometry
  + (size_t)(4 * PADN * STR) * sizeof(float);                 // Sre,Sim,Rre,Rim

__global__ __launch_bounds__(NTHREADS)
void mt2dte_kernel(const double* __restrict__ zn, const double* __restrict__ yn,
                   const double* __restrict__ freq, const double* __restrict__ ry,
                   const double* __restrict__ sig, const int* __restrict__ nzaP,
                   double* __restrict__ out, float2* __restrict__ wsC,
                   int nfreq, int nry)
{
  extern __shared__ unsigned char smem[];
  dc*     El   = (dc*)smem;            // 51
  dc*     Er   = El  + 51;             // 51
  dc*     exA  = Er  + 51;             // 41  : ex[nza, :]
  dc*     exB  = exA + 41;             // 41  : ex[nza+1, :]
  dc*     hfld = exB + 41;             // 41  : hys
  dc*     bAr  = hfld + 41;            // NN
  dc*     yAr  = bAr + NN;             // NN  (y during fwd, x after back-sub)
  double* dzs  = (double*)(yAr + NN);  // NZC
  double* dys  = dzs + NZC;            // NYC
  double* hzs  = dys + NYC;            // NZI
  double* hys_ = hzs + NZI;            // NYI
  float*  Sre  = (float*)(hys_ + NYI); // PADN*STR
  float*  Sim  = Sre + PADN*STR;
  float*  Rre  = Sim + PADN*STR;
  float*  Rim  = Rre + PADN*STR;

  const int tid  = threadIdx.x;
  const int wave = tid >> 5;
  const int lane = tid & 31;
  const int f    = blockIdx.x;

  const double omega  = 2.0 * PI_D * freq[f];
  const double om_miu = omega * MIU_D;
  float2* wsF = wsC + (size_t)f * (size_t)(NYI-1) * NZI * NZI;

  // -------- geometry --------
  for (int i = tid; i < NZC; i += NTHREADS) dzs[i] = zn[i+1] - zn[i];
  for (int i = tid; i < NYC; i += NTHREADS) dys[i] = yn[i+1] - yn[i];
  // warm L2/WGP$ with sigma while we are at it
  for (int i = tid; i < NZC*NYC; i += NTHREADS) __builtin_prefetch(&sig[i], 0, 1);
  __syncthreads();
  for (int i = tid; i < NZI; i += NTHREADS) hzs[i]  = 0.5*(dzs[i] + dzs[i+1]);
  for (int i = tid; i < NYI; i += NTHREADS) hys_[i] = 0.5*(dys[i] + dys[i+1]);
  __syncthreads();

  // -------- 1-D boundary solves (two lanes in different waves) --------
  if (tid == 0)  solve_1d_te(dzs, hzs, sig, 0,     om_miu, El);
  if (tid == 32) solve_1d_te(dzs, hzs, sig, NYC-1, om_miu, Er);
  __syncthreads();

  // -------- RHS vector b --------
  for (int idx = tid; idx < NN; idx += NTHREADS) {
    int m = idx / NZI, k = idx % NZI;
    double re = 0.0, im = 0.0;
    if (k == 0) re += -(hys_[m] / dzs[0]);
    if (m == 0)       { double c = -(hzs[k]/dys[0]);     re += c*El[k+1].re; im += c*El[k+1].im; }
    if (m == NYI-1)   { double c = -(hzs[k]/dys[NYC-1]); re += c*Er[k+1].re; im += c*Er[k+1].im; }
    bAr[idx] = {re, im};
  }
  __syncthreads();

  // =========================================================================
  // Forward block-Thomas sweep over the 39 y-columns.
  // =========================================================================
  for (int m = 0; m < NYI; ++m) {
    // ---- assemble S = D_m - V_m * C_{m-1}  (C_{m-1} lives in R from prev step)
    for (int idx = tid; idx < PADN*PADN; idx += NTHREADS) {
      int i = idx / PADN, k = idx % PADN;
      float sr = 0.f, si = 0.f;
      if (i < NZI && k < NZI) {
        double vr = 0.0, vi = 0.0;
        if (i == k) {
          double area = hzs[k] * hys_[m];
          double w00 = dzs[k]*dys[m],   w01 = dzs[k]*dys[m+1];
          double w10 = dzs[k+1]*dys[m], w11 = dzs[k+1]*dys[m+1];
          double sigc = (sig[k*NYC+m]*w00 + sig[k*NYC+m+1]*w01 +
                         sig[(k+1)*NYC+m]*w10 + sig[(k+1)*NYC+m+1]*w11) / (4.0*area);
          double val = hzs[k]*(1.0/dys[m] + 1.0/dys[m+1]) +
                       hys_[m]*(1.0/dzs[k] + 1.0/dzs[k+1]);
          vr = -val; vi = om_miu * sigc * area;
        } else if (k == i-1) vr = hys_[m] / dzs[i];
        else if   (k == i+1) vr = hys_[m] / dzs[k];
        if (m > 0) {
          double v = hzs[i] / dys[m];
          vr -= v * (double)Rre[i*STR + k];
          vi -= v * (double)Rim[i*STR + k];
        }
        sr = (float)vr; si = (float)vi;
      } else if (i == k) sr = 1.f;        // identity padding
      Sre[i*STR + k] = sr; Sim[i*STR + k] = si;
    }
    // ---- persist C_{m-1} for back-substitution (L2-resident scratch)
    if (m > 0) {
      float2* Cm = wsF + (size_t)(m-1) * NZI * NZI;
      for (int idx = tid; idx < NZI*NZI; idx += NTHREADS) {
        int i = idx / NZI, k = idx % NZI;
        Cm[idx] = make_float2(Rre[i*STR+k], Rim[i*STR+k]);
      }
    }
    __syncthreads();

    // ---- build RHS panel: cols 0..48 = diag(U_m), col 49 = b_m - V_m y_{m-1}
    for (int idx = tid; idx < PADN*PADN; idx += NTHREADS) {
      int i = idx / PADN, c = idx % PADN;
      float rr = 0.f, ri = 0.f;
      if (i < NZI) {
        if (c == i && m < NYI-1) {
          rr = (float)(hzs[i] / dys[m+1]);
        } else if (c == NZI) {
          dc r = bAr[m*NZI + i];
          if (m > 0) {
            double v = hzs[i] / dys[m];
            dc yp = yAr[(m-1)*NZI + i];
            r.re -= v*yp.re; r.im -= v*yp.im;
          }
          rr = (float)r.re; ri = (float)r.im;
        }
      }
      Rre[i*STR + c] = rr; Rim[i*STR + c] = ri;
    }
    __syncthreads();

    // ---- in-place complex blocked LU of S (right-looking, panel = 16,
    //      no pivoting; trailing updates on the WMMA pipe)
    for (int pb = 0; pb < 4; ++pb) {
      const int kb = 16*pb;
      // panel factor: columns kb..kb+15, rows kb..63 (pad rows are identity)
      for (int j = kb; j < kb+16; ++j) {
        float pr = Sre[j*STR+j], pi = Sim[j*STR+j];
        double den = 1.0 / ((double)pr*pr + (double)pi*pi);
        float ir = (float)( pr*den), ii = (float)(-pi*den);
        for (int i = j+1+tid; i < PADN; i += NTHREADS) {
          float ar = Sre[i*STR+j], ai = Sim[i*STR+j];
          Sre[i*STR+j] = ar*ir - ai*ii;
          Sim[i*STR+j] = ar*ii + ai*ir;
        }
        __syncthreads();
        int nr = PADN-1-j, nc = kb+15-j;      // within-panel trailing cols only
        for (int idx = tid; idx < nr*nc; idx += NTHREADS) {
          int i = j+1 + idx / nc, c = j+1 + idx % nc;
          float lr = Sre[i*STR+j], li = Sim[i*STR+j];
          float ur = Sre[j*STR+c], ui = Sim[j*STR+c];
          Sre[i*STR+c] -= lr*ur - li*ui;
          Sim[i*STR+c] -= lr*ui + li*ur;
        }
        __syncthreads();
      }
      const int c0col = kb + 16, nc2 = PADN - c0col;
      if (nc2 > 0) {
        // row TRSM: U rows kb..kb+15, cols kb+16..63  (apply unit-L panel)
        for (int r = kb; r < kb+16; ++r) {
          int ni = kb+15-r;
          for (int idx = tid; idx < ni*nc2; idx += NTHREADS) {
            int i = r+1 + idx / nc2, c = c0col + idx % nc2;
            float lr = Sre[i*STR+r], li = Sim[i*STR+r];
            float ur = Sre[r*STR+c], ui = Sim[r*STR+c];
            Sre[i*STR+c] -= lr*ur - li*ui;
            Sim[i*STR+c] -= lr*ui + li*ur;
          }
          __syncthreads();
        }
        // trailing update S[kb+16.., kb+16..] -= L_panel * U_panel  (WMMA)
        const int nt = 3 - pb;                 // 16x16 tiles per side
        for (int t = wave; t < nt*nt; t += 8) {  // wave-uniform => EXEC all 1s
          int qt = pb+1 + t / nt, jt = pb+1 + t % nt;
          cgemm16_sub(Sre, Sim, 16*qt, kb, Sre, Sim, kb, 16*jt,
                      Sre, Sim, 16*qt, 16*jt, lane);
        }
        __syncthreads();
      }
    }

    // ---- blocked TRSM forward: R <- L^{-1} R  (unit lower)
    for (int p = 0; p < 4; ++p) {
      int r0 = 16*p;
      for (int r = r0; r < r0+16; ++r) {
        int ni = r0+16-1-r;
        for (int idx = tid; idx < ni*PADN; idx += NTHREADS) {
          int i = r+1 + idx / PADN, c = idx % PADN;
          float lr = Sre[i*STR+r], li = Sim[i*STR+r];
          float xr = Rre[r*STR+c], xi = Rim[r*STR+c];
          Rre[i*STR+c] -= lr*xr - li*xi;
          Rim[i*STR+c] -= lr*xi + li*xr;
        }
        __syncthreads();
      }
      int ntile = (3-p) * 4;                       // panels below x 4 col-tiles
      for (int t = wave; t < ntile; t += 8) {      // wave-uniform => EXEC all 1s
        int q = p+1 + t/4, J0 = 16*(t & 3);
        cgemm16_sub(Sre, Sim, 16*q, 16*p, Rre, Rim, 16*p, J0,
                    Rre, Rim, 16*q, J0, lane);
      }
      __syncthreads();
    }

    // ---- blocked TRSM backward: R <- U^{-1} R
    for (int p = 3; p >= 0; --p) {
      int r0 = 16*p;
      for (int r = r0+15; r >= r0; --r) {
        if (r < NZI) {                              // padded rows have U[r][r]=1
          float pr = Sre[r*STR+r], pi = Sim[r*STR+r];
          double den = 1.0 / ((double)pr*pr + (double)pi*pi);
          float ir = (float)( pr*den), ii = (float)(-pi*den);
          for (int c = tid; c < PADN; c += NTHREADS) {
            float xr = Rre[r*STR+c], xi = Rim[r*STR+c];
            Rre[r*STR+c] = xr*ir - xi*ii;
            Rim[r*STR+c] = xr*ii + xi*ir;
          }
        }
        __syncthreads();
        int ni = r - r0;
        for (int idx = tid; idx < ni*PADN; idx += NTHREADS) {
          int i = r0 + idx / PADN, c = idx % PADN;
          float ur = Sre[i*STR+r], ui = Sim[i*STR+r];
          float xr = Rre[r*STR+c], xi = Rim[r*STR+c];
          Rre[i*STR+c] -= ur*xr - ui*xi;
          Rim[i*STR+c] -= ur*xi + ui*xr;
        }
        __syncthreads();
      }
      int ntile = p * 4;
      for (int t = wave; t < ntile; t += 8) {
        int q = t/4, J0 = 16*(t & 3);
        cgemm16_sub(Sre, Sim, 16*q, 16*p, Rre, Rim, 16*p, J0,
                    Rre, Rim, 16*q, J0, lane);
      }
      __syncthreads();
    }

    // ---- extract y_m (column 49); C_m (cols 0..48) stays in R for next step
    for (int i = tid; i < NZI; i += NTHREADS) {
      yAr[m*NZI + i].re = (double)Rre[i*STR + NZI];
      yAr[m*NZI + i].im = (double)Rim[i*STR + NZI];
    }
    __syncthreads();
  }

  // =========================================================================
  // Back substitution: x_m = y_m - C_m x_{m+1}   (f64 accumulate, f32 C)
  // =========================================================================
  for (int m = NYI-2; m >= 0; --m) {
    const float2* Cm = wsF + (size_t)m * NZI * NZI;
    for (int i = tid; i < NZI; i += NTHREADS) {
      double ar = yAr[m*NZI + i].re, ai = yAr[m*NZI + i].im;
      for (int k = 0; k < NZI; ++k) {
        float2 cc = Cm[i*NZI + k];
        double xr = yAr[(m+1)*NZI + k].re, xi = yAr[(m+1)*NZI + k].im;
        ar -= (double)cc.x*xr - (double)cc.y*xi;
        ai -= (double)cc.x*xi + (double)cc.y*xr;
      }
      yAr[m*NZI + i] = {ar, ai};
    }
    __syncthreads();
  }

  // =========================================================================
  // Receiver post-processing
  // =========================================================================
  const int nzaV = nzaP[0];
  for (int n = tid; n <= NYC; n += NTHREADS) {
    // ex rows nza and nza+1 (row 0 = 1+0j, bottom row = 0, sides = El/Er)
    for (int s = 0; s < 2; ++s) {
      int rz = nzaV + s;
      dc v;
      if (rz == 0)            v = {1.0, 0.0};
      else if (rz >= NZC)     v = {0.0, 0.0};
      else if (n == 0)        v = El[rz];
      else if (n == NYC)      v = Er[rz];
      else                    v = yAr[(n-1)*NZI + (rz-1)];
      (s == 0 ? exA : exB)[n] = v;
    }
  }
  __syncthreads();
  for (int n = tid; n <= NYC; n += NTHREADS) {
    double delz = dzs[nzaV];
    double sn;
    if (n == 0)        sn = sig[nzaV*NYC + 0];
    else if (n == NYC) sn = sig[nzaV*NYC + NYC-1];
    else               sn = 0.5*(sig[nzaV*NYC + n-1] + sig[nzaV*NYC + n]);
    double t1im = -1.0 / (om_miu * delz);       // t1 = -i/(omega*miu*delz)
    dc c0 = { 0.375*sn*delz, -t1im };           // -t1 + 0.375 sn dz
    dc c1 = { 0.125*sn*delz,  t1im };           //  t1 + 0.125 sn dz
    hfld[n] = cadd(cmul(c0, exA[n]), cmul(c1, exB[n]));
  }
  __syncthreads();
  for (int r = tid; r < nry; r += NTHREADS) {
    double v = ry[r];
    int idx = 0;
    while (idx <= NYC && yn[idx] < v) ++idx;    // searchsorted(side='left')
    if (idx < 1)  idx = 1;
    if (idx > NYC) idx = NYC;
    double xl = yn[idx-1], xr = yn[idx];
    double w = (v - xl) / (xr - xl + 1e-12);
    dc exr = { exA[idx-1].re + w*(exA[idx].re - exA[idx-1].re),
               exA[idx-1].im + w*(exA[idx].im - exA[idx-1].im) };
    dc hyr = { hfld[idx-1].re + w*(hfld[idx].re - hfld[idx-1].re),
               hfld[idx-1].im + w*(hfld[idx].im - hfld[idx-1].im) };
    dc Z = cdiv(exr, hyr);
    double rho = (Z.re*Z.re + Z.im*Z.im) / om_miu;
    double phs = -atan2(Z.im, Z.re) * (180.0 / PI_D);
    out[f*nry + r]              = rho;
    out[nfreq*nry + f*nry + r]  = phs;
    double* zo = out + 2*(size_t)nfreq*nry;
    zo[2*((size_t)f*nry + r) + 0] = Z.re;
    zo[2*((size_t)f*nry + r) + 1] = Z.im;
  }
}

// ---------------------------------------------------------------------------
extern "C" void kernel_launch(void* const* d_in, const int* in_sizes, int n_in,
                              void* d_out, int out_size, void* d_ws, size_t ws_size,
                              hipStream_t stream) {
  const double* zn   = (const double*)d_in[0];   // 51
  const double* yn   = (const double*)d_in[1];   // 41
  const double* freq = (const double*)d_in[2];   // 2
  const double* ry   = (const double*)d_in[3];   // 30
  const double* sig  = (const double*)d_in[4];   // 50*40
  const int*    nza  = (const int*)d_in[5];      // 1
  double* out = (double*)d_out;

  const int nfreq = in_sizes[2];
  const int nry   = in_sizes[3];

  // d_ws: per-frequency Schur blocks C_m: nfreq * 38 * 49 * 49 * sizeof(float2)
  // (~1.46 MB total for 2 freqs) - L2 resident on the 192 MB L2.
  mt2dte_kernel<<<dim3(nfreq), dim3(NTHREADS), SMEM_BYTES, stream>>>(
      zn, yn, freq, ry, sig, nza, out, (float2*)d_ws, nfreq, nry);
}